// Decoder_GRU_38250978738642
// MI455X (gfx1250) — compile-verified
//
#include <hip/hip_runtime.h>

// ---------------------------------------------------------------------------
// CDNA5 (gfx1250) fp32 WMMA pipeline for the Chebyshev graph-conv GRU.
// Wave32; V_WMMA_F32_16X16X4_F32 throughout. L-apply GEMM stages the shared
// X operand in LDS, pair-swizzled so B fragments come straight out of
// ds_load_b128 destination registers (no shuffle movs).
// ---------------------------------------------------------------------------

typedef __attribute__((ext_vector_type(2))) float v2f;
typedef __attribute__((ext_vector_type(4))) float v4f;
typedef __attribute__((ext_vector_type(8))) float v8f;

#define WMMA_F32_16x16x4(A, Bf, C) \
  __builtin_amdgcn_wmma_f32_16x16x4_f32(false, (A), false, (Bf), (short)0, (C), false, false)

__device__ __forceinline__ float dev_sigmoid(float x) {
  return 1.0f / (1.0f + __expf(-x));
}

// ---------------------------------------------------------------------------
// Y[b, n, 0:64] = alpha * sum_m L[n, m] * X[b, m, 0:64] + beta * Z[b, n, 0:64]
// L: (M, M) row-major fp32. X, Y, Z: (B, M, 64).
// grid = (M/64, B), block = 128 (4 waves). Each wave owns a 16x64 output
// strip (4 WMMA accumulators); the workgroup covers 64 rows and shares the
// X K-chunk via LDS.
//
// LDS pair-swizzle: X[k][n] stored at
//   (k>>1)*128 + (n&15)*8 + (n>>4)*2 + (k&1)
// so a lane's 8 B-values for one k-pair are contiguous:
//   [t0k0 t0k1 t1k0 t1k1 | t2k0 t2k1 t3k0 t3k1]
// -> two ds_load_b128, each holding two even-aligned v2f WMMA B operands.
//
// ISA fragment layouts (CDNA5 ISA):
//   A (16x4 f32, 2 VGPR): lane l -> row = l&15, half = l>>4; vgpr v holds
//     K = half*2 + v (contiguous float2 in row-major L).
//   B (4x16 f32, 2 VGPR): lane l -> col n = l&15; vgpr v holds K = half*2+v.
//   C/D (16x16 f32, 8 VGPR): lane l -> n = l&15; vgpr v holds M = v + 8*half.
// ---------------------------------------------------------------------------
#define LX_KC 32

__global__ __launch_bounds__(128) void lx_gemm(
    const float* __restrict__ L, const float* __restrict__ X,
    const float* __restrict__ Z, float* __restrict__ Y,
    int M, float alpha, float beta) {
  __shared__ float sX[LX_KC * 64];

  const int tid  = threadIdx.x;
  const int wave = tid >> 5;
  const int lane = tid & 31;
  const int lidx = lane & 15;   // row of A / col of B / col of D
  const int kh   = lane >> 4;   // K-half selector
  const int nbase = blockIdx.x * 64 + wave * 16;
  const int b     = blockIdx.y;

  const float* Lrow = L + (size_t)(nbase + lidx) * (size_t)M;
  const float* Xb   = X + (size_t)b * (size_t)M * 64;

  v8f acc0 = {}, acc1 = {}, acc2 = {}, acc3 = {};

  for (int k0 = 0; k0 < M; k0 += LX_KC) {
    __syncthreads();  // previous chunk's LDS reads complete before overwrite
    // Stage X[k0:k0+KC, 0:64] into LDS, pair-swizzled. 128 thr x 4 float4.
#pragma unroll
    for (int i = tid; i < LX_KC * 16; i += 128) {
      const int kk = i >> 4;
      const int c0 = (i & 15) * 4;
      const float4 v = *(const float4*)(Xb + (size_t)(k0 + kk) * 64 + c0);
      float* dst = sX + (kk >> 1) * 128 + (c0 & 15) * 8 + (c0 >> 4) * 2 + (kk & 1);
      dst[0]  = v.x;
      dst[8]  = v.y;
      dst[16] = v.z;
      dst[24] = v.w;
    }
    __syncthreads();

    // Prefetch next A chunk (L is L2-resident; speculative, drop on fault).
    __builtin_prefetch(Lrow + k0 + LX_KC, 0, 1);

#pragma unroll
    for (int k = 0; k < LX_KC; k += 4) {
      // A fragment: L[nbase+lidx][k0 + k + kh*2 + {0,1}]
      const v2f a = *(const v2f*)(Lrow + k0 + k + kh * 2);

      // B fragments: lane's k-pair = k/2 + kh; 8 contiguous floats.
      const float* bp = sX + ((k >> 1) + kh) * 128 + lidx * 8;
      const v4f q0 = *(const v4f*)(bp);       // [b0.x b0.y b1.x b1.y]
      const v4f q1 = *(const v4f*)(bp + 4);   // [b2.x b2.y b3.x b3.y]
      v2f b0, b1, b2, b3;
      b0.x = q0.x; b0.y = q0.y;
      b1.x = q0.z; b1.y = q0.w;
      b2.x = q1.x; b2.y = q1.y;
      b3.x = q1.z; b3.y = q1.w;

      acc0 = WMMA_F32_16x16x4(a, b0, acc0);
      acc1 = WMMA_F32_16x16x4(a, b1, acc1);
      acc2 = WMMA_F32_16x16x4(a, b2, acc2);
      acc3 = WMMA_F32_16x16x4(a, b3, acc3);
    }
  }

  float* Yb = Y + (size_t)b * (size_t)M * 64;
  const float* Zb = Z ? (Z + (size_t)b * (size_t)M * 64) : nullptr;

#pragma unroll
  for (int v = 0; v < 8; ++v) {
    const int r = v + kh * 8;
    const size_t base = (size_t)(nbase + r) * 64 + lidx;
    float z0 = 0.f, z1 = 0.f, z2 = 0.f, z3 = 0.f;
    if (Zb) { z0 = Zb[base]; z1 = Zb[base + 16]; z2 = Zb[base + 32]; z3 = Zb[base + 48]; }
    Yb[base]      = alpha * acc0[v] + beta * z0;
    Yb[base + 16] = alpha * acc1[v] + beta * z1;
    Yb[base + 32] = alpha * acc2[v] + beta * z2;
    Yb[base + 48] = alpha * acc3[v] + beta * z3;
  }
}

// ---------------------------------------------------------------------------
// out[r, 0:FOUT] = act( sum_s T_s[r, 0:64] @ W[s, 0:64, 0:FOUT] )
// T0/T1/T2: (Rows, 64). W: (3, 64, FOUT). ACT: 0 = sigmoid, 1 = tanh.
// grid = Rows/16, block = 32.
// ---------------------------------------------------------------------------
template <int FOUT, int ACT>
__global__ __launch_bounds__(32) void feat3_gemm(
    const float* __restrict__ T0, const float* __restrict__ T1,
    const float* __restrict__ T2, const float* __restrict__ W,
    float* __restrict__ out) {
  const int lane = threadIdx.x;
  const int lidx = lane & 15;
  const int kh   = lane >> 4;
  const int rbase = blockIdx.x * 16;
  constexpr int NT = FOUT / 16;

  v8f acc[NT] = {};
  const float* Ts[3] = {T0, T1, T2};

  for (int s = 0; s < 3; ++s) {
    const float* T  = Ts[s];
    const float* Ws = W + s * 64 * FOUT;
#pragma unroll
    for (int k = 0; k < 64; k += 4) {
      const v2f a = *(const v2f*)(T + (size_t)(rbase + lidx) * 64 + k + kh * 2);
      const float* wrow = Ws + (k + kh * 2) * FOUT + lidx;
#pragma unroll
      for (int t = 0; t < NT; ++t) {
        v2f bb;
        bb.x = wrow[t * 16];
        bb.y = wrow[FOUT + t * 16];
        acc[t] = WMMA_F32_16x16x4(a, bb, acc[t]);
      }
    }
  }

#pragma unroll
  for (int v = 0; v < 8; ++v) {
    const int r = v + kh * 8;
#pragma unroll
    for (int t = 0; t < NT; ++t) {
      float val = acc[t][v];
      val = (ACT == 0) ? dev_sigmoid(val) : tanhf(val);
      out[(size_t)(rbase + r) * FOUT + t * 16 + lidx] = val;
    }
  }
}

// ---------------------------------------------------------------------------
// y[r, 0:32] = sigmoid( H[r, 0:32] @ W(32x32) + bias )   grid = Rows/16
// ---------------------------------------------------------------------------
__global__ __launch_bounds__(32) void head_gemm(
    const float* __restrict__ H, const float* __restrict__ W,
    const float* __restrict__ bias, float* __restrict__ out) {
  const int lane = threadIdx.x;
  const int lidx = lane & 15;
  const int kh   = lane >> 4;
  const int rbase = blockIdx.x * 16;

  v8f acc0 = {}, acc1 = {};
#pragma unroll
  for (int k = 0; k < 32; k += 4) {
    const v2f a = *(const v2f*)(H + (size_t)(rbase + lidx) * 32 + k + kh * 2);
    const float* wrow = W + (k + kh * 2) * 32 + lidx;
    v2f b0, b1;
    b0.x = wrow[0];   b0.y = wrow[32 + 0];
    b1.x = wrow[16];  b1.y = wrow[32 + 16];
    acc0 = WMMA_F32_16x16x4(a, b0, acc0);
    acc1 = WMMA_F32_16x16x4(a, b1, acc1);
  }

  const float bias0 = bias[lidx];
  const float bias1 = bias[16 + lidx];
#pragma unroll
  for (int v = 0; v < 8; ++v) {
    const int r = v + kh * 8;
    out[(size_t)(rbase + r) * 32 + lidx]      = dev_sigmoid(acc0[v] + bias0);
    out[(size_t)(rbase + r) * 32 + 16 + lidx] = dev_sigmoid(acc1[v] + bias1);
  }
}

// ---------------------------------------------------------------------------
// dst[row, 0:32] = x[row, 0:32];  dst[row, 32:64] = (gates ? r*h : h)
// gates holds post-sigmoid values; r = gates[row, 0:32]. total = Rows*32.
// ---------------------------------------------------------------------------
__global__ void concat_kernel(const float* __restrict__ x,
                              const float* __restrict__ h,
                              const float* __restrict__ gates,
                              float* __restrict__ dst, int total) {
  const int i = blockIdx.x * blockDim.x + threadIdx.x;
  if (i >= total) return;
  const int row = i >> 5;
  const int f   = i & 31;
  float hv = h[i];
  if (gates) hv *= gates[(size_t)row * 64 + f];
  dst[(size_t)row * 64 + f]      = x[i];
  dst[(size_t)row * 64 + 32 + f] = hv;
}

// h[i] = u*h[i] + (1-u)*c[i]   with u = gates[row, 32+f]   (in-place state)
__global__ void hy_kernel(const float* __restrict__ gates,
                          const float* __restrict__ c,
                          float* __restrict__ h, int total) {
  const int i = blockIdx.x * blockDim.x + threadIdx.x;
  if (i >= total) return;
  const int row = i >> 5;
  const int f   = i & 31;
  const float u = gates[(size_t)row * 64 + 32 + f];
  h[i] = u * h[i] + (1.0f - u) * c[i];
}

// ---------------------------------------------------------------------------
// Host: one GRU step for one graph (node or edge).
// ---------------------------------------------------------------------------
static void run_graph_step(const float* x, const float* L, const float* Wg,
                           const float* Wu, const float* Wh, const float* bh,
                           float* h, float* comb, float* t1, float* t2,
                           float* g, float* c, float* out_t, int M,
                           hipStream_t stream) {
  const int rows = 4 * M;        // B * M
  const int totF = rows * 32;
  const dim3 blkL(128);
  const dim3 gridL(M / 64, 4);   // (64-row tiles, batch)
  const dim3 blkW(32);
  const int eb = 256;
  const int egrid = (totF + eb - 1) / eb;

  // gates path: comb = [x, h]; T1 = L comb; T2 = 2 L T1 - comb
  concat_kernel<<<egrid, eb, 0, stream>>>(x, h, nullptr, comb, totF);
  lx_gemm<<<gridL, blkL, 0, stream>>>(L, comb, nullptr, t1, M, 1.0f, 0.0f);
  lx_gemm<<<gridL, blkL, 0, stream>>>(L, t1, comb, t2, M, 2.0f, -1.0f);
  feat3_gemm<64, 0><<<rows / 16, blkW, 0, stream>>>(comb, t1, t2, Wg, g);

  // candidate path: comb = [x, r*h]
  concat_kernel<<<egrid, eb, 0, stream>>>(x, h, g, comb, totF);
  lx_gemm<<<gridL, blkL, 0, stream>>>(L, comb, nullptr, t1, M, 1.0f, 0.0f);
  lx_gemm<<<gridL, blkL, 0, stream>>>(L, t1, comb, t2, M, 2.0f, -1.0f);
  feat3_gemm<32, 1><<<rows / 16, blkW, 0, stream>>>(comb, t1, t2, Wu, c);

  // state update + output head
  hy_kernel<<<egrid, eb, 0, stream>>>(g, c, h, totF);
  head_gemm<<<rows / 16, blkW, 0, stream>>>(h, Wh, bh, out_t);
}

extern "C" void kernel_launch(void* const* d_in, const int* in_sizes, int n_in,
                              void* d_out, int out_size, void* d_ws, size_t ws_size,
                              hipStream_t stream) {
  (void)in_sizes; (void)n_in; (void)out_size; (void)ws_size;

  const float* x_n  = (const float*)d_in[0];   // (4,1024,32)
  const float* x_e  = (const float*)d_in[1];   // (4,4096,32)
  const float* L_n  = (const float*)d_in[2];   // (1024,1024)
  const float* L_e  = (const float*)d_in[3];   // (4096,4096)
  const float* Wg_n = (const float*)d_in[4];   // (3,64,64)
  const float* Wg_e = (const float*)d_in[5];   // (3,64,64)
  const float* Wu_n = (const float*)d_in[6];   // (3,64,32)
  const float* Wu_e = (const float*)d_in[7];   // (3,64,32)
  const float* W_n  = (const float*)d_in[8];   // (32,32)
  const float* b_n  = (const float*)d_in[9];   // (32)
  const float* W_e  = (const float*)d_in[10];  // (32,32)
  const float* b_e  = (const float*)d_in[11];  // (32)
  float* out = (float*)d_out;

  const int T = 6, B = 4, N = 1024, E = 4096, F = 32;

  // ---- workspace carve-out (fp32), ~25 MB total ----
  float* ws = (float*)d_ws;
  float* h_n    = ws; ws += B * N * F;
  float* comb_n = ws; ws += B * N * 64;
  float* t1_n   = ws; ws += B * N * 64;
  float* t2_n   = ws; ws += B * N * 64;
  float* g_n    = ws; ws += B * N * 64;
  float* c_n    = ws; ws += B * N * F;
  float* h_e    = ws; ws += B * E * F;
  float* comb_e = ws; ws += B * E * 64;
  float* t1_e   = ws; ws += B * E * 64;
  float* t2_e   = ws; ws += B * E * 64;
  float* g_e    = ws; ws += B * E * 64;
  float* c_e    = ws; ws += B * E * F;

  hipMemsetAsync(h_n, 0, (size_t)B * N * F * sizeof(float), stream);
  hipMemsetAsync(h_e, 0, (size_t)B * E * F * sizeof(float), stream);

  float* out_node = out;                          // (T,B,N,F)
  float* out_edge = out + (size_t)T * B * N * F;  // (T,B,E,F)

  for (int t = 0; t < T; ++t) {
    run_graph_step(x_n, L_n, Wg_n, Wu_n, W_n, b_n,
                   h_n, comb_n, t1_n, t2_n, g_n, c_n,
                   out_node + (size_t)t * B * N * F, N, stream);
    run_graph_step(x_e, L_e, Wg_e, Wu_e, W_e, b_e,
                   h_e, comb_e, t1_e, t2_e, g_e, c_e,
                   out_edge + (size_t)t * B * E * F, E, stream);
  }
}